// MultiHeadAdditiveAttention_81724637708862
// MI455X (gfx1250) — compile-verified
//
#include <hip/hip_runtime.h>
#include <hip/hip_bf16.h>

// Problem constants (B=4, S=512, D=256, H=1)
#define BB 4
#define SS 512
#define DD 256
#define BS (BB * SS)   // 2048 rows total for the projection GEMMs

#define KPITCH 260     // LDS row pitch (floats): 1040B row stride -> 16B aligned,
                       // bank = (4*ki + d) % 64 distinct for ki in 0..15

typedef __attribute__((ext_vector_type(2))) float v2f;
typedef __attribute__((ext_vector_type(8))) float v8f;

// ---------------------------------------------------------------------------
// fp32 WMMA 16x16x4 wrapper (D = A(16x4) * B(4x16) + C), wave32.
// A layout: lane l -> row M = l&15; VGPR j holds K = j + 2*(l>>4)
// B layout: lane l -> col N = l&15; VGPR j holds K = j + 2*(l>>4)
// C/D     : lane l -> col N = l&15; VGPR v holds row M = v + 8*(l>>4)
// ---------------------------------------------------------------------------
__device__ inline v8f wmma_k4(v2f a, v2f b, v8f c) {
#if __has_builtin(__builtin_amdgcn_wmma_f32_16x16x4_f32)
  return __builtin_amdgcn_wmma_f32_16x16x4_f32(
      /*neg_a=*/false, a, /*neg_b=*/false, b,
      /*c_mod=*/(short)0, c, /*reuse_a=*/false, /*reuse_b=*/false);
#else
  c[0] += a[0] * b[0] + a[1] * b[1];
  return c;
#endif
}

// Fast tanh: native v_tanh_f32 on gfx1250 if declared, else exp2+rcp form.
__device__ inline float fast_tanh(float x) {
#if __has_builtin(__builtin_amdgcn_tanhf)
  return __builtin_amdgcn_tanhf(x);
#else
  const float TWO_LOG2E = 2.885390081777927f;  // 2*log2(e)
  float xc = fminf(fmaxf(x, -15.0f), 15.0f);
  float t = __builtin_amdgcn_exp2f(xc * TWO_LOG2E);
#if __has_builtin(__builtin_amdgcn_rcpf)
  return (t - 1.0f) * __builtin_amdgcn_rcpf(t + 1.0f);
#else
  return (t - 1.0f) / (t + 1.0f);
#endif
#endif
}

// Async global -> LDS copy, 16 bytes per lane, tracked by ASYNCcnt.
// ldsOff: per-lane LDS byte offset (low 32 bits of the generic shared pointer),
// g: per-lane global source (16B aligned).
__device__ inline void async_copy_b128(unsigned ldsOff, const float* g) {
  asm volatile("global_load_async_to_lds_b128 %0, %1, off"
               :: "v"(ldsOff), "v"((unsigned long long)(uintptr_t)g)
               : "memory");
}
__device__ inline void wait_asynccnt0() {
  asm volatile("s_wait_asynccnt 0x0" ::: "memory");
}

// ---------------------------------------------------------------------------
// 16x16 output tile of  O = X * W^T + bias, K = 256, all row-major fp32.
// One wave (32 threads) per tile.
// ---------------------------------------------------------------------------
__device__ inline void gemm16x16_nt_k256(const float* __restrict__ X,
                                         const float* __restrict__ W,
                                         const float* __restrict__ bias,
                                         float* __restrict__ O,
                                         int m0, int n0) {
  const int lane = threadIdx.x & 31;
  const int r  = lane & 15;   // A-row / B-col within tile
  const int hv = lane >> 4;   // half-wave select
  const float* xrow = X + (size_t)(m0 + r) * DD;
  const float* wrow = W + (size_t)(n0 + r) * DD;  // W^T load: B[k][n] = W[n][k]
  v8f acc = {};
#pragma unroll 4
  for (int k = 0; k < DD; k += 4) {
    v2f a = *(const v2f*)(xrow + k + 2 * hv);
    v2f b = *(const v2f*)(wrow + k + 2 * hv);
    acc = wmma_k4(a, b, acc);
  }
  const float bn = bias[n0 + r];
#pragma unroll
  for (int i = 0; i < 8; ++i) {
    O[(size_t)(m0 + i + 8 * hv) * DD + (n0 + r)] = acc[i] + bn;
  }
}

// Kernel 1: fused Q/K/V projections. grid = (D/16, BS/16, 3), block = 32.
__global__ void k_proj_qkv(const float* __restrict__ q_in,
                           const float* __restrict__ k_in,
                           const float* __restrict__ v_in,
                           const float* __restrict__ Wq, const float* __restrict__ bq,
                           const float* __restrict__ Wk, const float* __restrict__ bk,
                           const float* __restrict__ Wv, const float* __restrict__ bv,
                           float* __restrict__ Q, float* __restrict__ Kp,
                           float* __restrict__ Vp) {
  const int which = blockIdx.z;
  const float* X = (which == 0) ? q_in : (which == 1) ? k_in : v_in;
  const float* W = (which == 0) ? Wq   : (which == 1) ? Wk   : Wv;
  const float* bb = (which == 0) ? bq  : (which == 1) ? bk   : bv;
  float* O = (which == 0) ? Q : (which == 1) ? Kp : Vp;
  gemm16x16_nt_k256(X, W, bb, O, blockIdx.y * 16, blockIdx.x * 16);
}

// Kernel 2: Bahdanau scores. grid = (S/16, S/16, B), block = 256.
// scores[b,q,k] = sum_d Ws[d] * tanh(Q[b,q,d] + K[b,k,d]) + bs
// q/k tiles staged via GLOBAL_LOAD_ASYNC_TO_LDS_B128 (ASYNCcnt-tracked).
__global__ void k_scores(const float* __restrict__ Q,
                         const float* __restrict__ Kp,
                         const float* __restrict__ Wsv,
                         const float* __restrict__ bsv,
                         float* __restrict__ attn) {
  __shared__ __align__(16) float qs[16][KPITCH];
  __shared__ __align__(16) float ks[16][KPITCH];
  __shared__ float wsh[DD];
  const int t  = threadIdx.x;          // 0..255
  const int b  = blockIdx.z;
  const int q0 = blockIdx.y * 16;
  const int k0 = blockIdx.x * 16;
  const float* Qb = Q  + ((size_t)b * SS + q0) * DD;
  const float* Kb = Kp + ((size_t)b * SS + k0) * DD;

  wsh[t] = Wsv[t];

  // 16 rows x 256 floats per tile; 64 lanes/row x 16B = one row per 64 threads;
  // 4 passes cover 16 rows for each of q and k.
  const int rowId = t >> 6;          // 0..3
  const int col   = (t & 63) << 2;   // 0,4,...,252 (16B-aligned lane chunk)
#pragma unroll
  for (int pass = 0; pass < 4; ++pass) {
    const int row = pass * 4 + rowId;
    async_copy_b128((unsigned)(uintptr_t)(&qs[row][col]), Qb + row * DD + col);
    async_copy_b128((unsigned)(uintptr_t)(&ks[row][col]), Kb + row * DD + col);
  }
  wait_asynccnt0();
  __syncthreads();

  const int qi = t >> 4;
  const int ki = t & 15;
  float acc = 0.0f;
#pragma unroll 4
  for (int d = 0; d < DD; ++d) {
    acc += wsh[d] * fast_tanh(qs[qi][d] + ks[ki][d]);
  }
  attn[((size_t)b * SS + (q0 + qi)) * SS + (k0 + ki)] = acc + bsv[0];
}

// Kernel 3: in-place softmax over rows of length 512. grid = B*S, block = 256.
__global__ void k_softmax512(float* __restrict__ attn) {
  __shared__ float red[256];
  const int t = threadIdx.x;
  float* p = attn + (size_t)blockIdx.x * SS;
  float a0 = p[t];
  float a1 = p[t + 256];
  red[t] = fmaxf(a0, a1);
  __syncthreads();
  for (int s = 128; s > 0; s >>= 1) {
    if (t < s) red[t] = fmaxf(red[t], red[t + s]);
    __syncthreads();
  }
  const float m = red[0];
  __syncthreads();
  const float LOG2E = 1.4426950408889634f;
  float e0 = __builtin_amdgcn_exp2f((a0 - m) * LOG2E);
  float e1 = __builtin_amdgcn_exp2f((a1 - m) * LOG2E);
  red[t] = e0 + e1;
  __syncthreads();
  for (int s = 128; s > 0; s >>= 1) {
    if (t < s) red[t] += red[t + s];
    __syncthreads();
  }
#if __has_builtin(__builtin_amdgcn_rcpf)
  const float inv = __builtin_amdgcn_rcpf(red[0]);
#else
  const float inv = 1.0f / red[0];
#endif
  p[t]       = e0 * inv;
  p[t + 256] = e1 * inv;
}

// Kernel 4: attended = attn @ V (batched). grid = (D/16, S/16, B), block = 32.
__global__ void k_attend(const float* __restrict__ attn,
                         const float* __restrict__ V,
                         float* __restrict__ att_out) {
  const int b  = blockIdx.z;
  const int m0 = blockIdx.y * 16;
  const int n0 = blockIdx.x * 16;
  const int lane = threadIdx.x & 31;
  const int r  = lane & 15;
  const int hv = lane >> 4;
  const float* A    = attn + (size_t)b * SS * SS;
  const float* Vb   = V    + (size_t)b * SS * DD;
  const float* arow = A + (size_t)(m0 + r) * SS;
  v8f acc = {};
  for (int k = 0; k < SS; k += 4) {
    v2f a = *(const v2f*)(arow + k + 2 * hv);
    const int kr = k + 2 * hv;
    v2f bvec = { Vb[(size_t)kr * DD + n0 + r],
                 Vb[(size_t)(kr + 1) * DD + n0 + r] };
    acc = wmma_k4(a, bvec, acc);
  }
#pragma unroll
  for (int i = 0; i < 8; ++i) {
    att_out[((size_t)b * SS + m0 + i + 8 * hv) * DD + (n0 + r)] = acc[i];
  }
}

// Kernel 5: out = attended @ Wo^T + bo. grid = (D/16, BS/16), block = 32.
__global__ void k_out_proj(const float* __restrict__ A,
                           const float* __restrict__ Wo,
                           const float* __restrict__ bo,
                           float* __restrict__ out) {
  gemm16x16_nt_k256(A, Wo, bo, out, blockIdx.y * 16, blockIdx.x * 16);
}

// ---------------------------------------------------------------------------
// Launch: inputs (setup_inputs order):
//  0 query, 1 key, 2 value, 3 Wq, 4 bq, 5 Wk, 6 bk, 7 Wv, 8 bv,
//  9 Ws, 10 bs, 11 Wo, 12 bo
// d_out = [out (B*S*D) | attn (B*1*S*S)] fp32
// d_ws  = [Q | K | V | attended] fp32  (4 * 2 MB = 8 MB)
// ---------------------------------------------------------------------------
extern "C" void kernel_launch(void* const* d_in, const int* in_sizes, int n_in,
                              void* d_out, int out_size, void* d_ws, size_t ws_size,
                              hipStream_t stream) {
  (void)in_sizes; (void)n_in; (void)out_size; (void)ws_size;

  const float* query = (const float*)d_in[0];
  const float* key   = (const float*)d_in[1];
  const float* value = (const float*)d_in[2];
  const float* Wq = (const float*)d_in[3];
  const float* bq = (const float*)d_in[4];
  const float* Wk = (const float*)d_in[5];
  const float* bk = (const float*)d_in[6];
  const float* Wv = (const float*)d_in[7];
  const float* bv = (const float*)d_in[8];
  const float* Ws = (const float*)d_in[9];
  const float* bs = (const float*)d_in[10];
  const float* Wo = (const float*)d_in[11];
  const float* bo = (const float*)d_in[12];

  float* out  = (float*)d_out;              // (B, S, D)
  float* attn = out + (size_t)BS * DD;      // (B, 1, S, S)

  float* ws   = (float*)d_ws;
  float* Q    = ws;                          // (BS, D)
  float* Kp   = Q  + (size_t)BS * DD;        // (BS, D)
  float* Vp   = Kp + (size_t)BS * DD;        // (BS, D)
  float* Att  = Vp + (size_t)BS * DD;        // (BS, D) attended

  // 1) Q/K/V projections (WMMA fp32)
  k_proj_qkv<<<dim3(DD / 16, BS / 16, 3), 32, 0, stream>>>(
      query, key, value, Wq, bq, Wk, bk, Wv, bv, Q, Kp, Vp);

  // 2) additive scores -> raw attn logits (transcendental-bound stage,
  //    async-to-LDS staging)
  k_scores<<<dim3(SS / 16, SS / 16, BB), 256, 0, stream>>>(Q, Kp, Ws, bs, attn);

  // 3) softmax in place (attn is also the second output)
  k_softmax512<<<dim3(BS), 256, 0, stream>>>(attn);

  // 4) attended = attn @ V (WMMA fp32)
  k_attend<<<dim3(DD / 16, SS / 16, BB), 32, 0, stream>>>(attn, Vp, Att);

  // 5) output projection (WMMA fp32)
  k_out_proj<<<dim3(DD / 16, BS / 16, 1), 32, 0, stream>>>(Att, Wo, bo, out);
}